// PsRoiOffset_69879117906603
// MI455X (gfx1250) — compile-verified
//
#include <hip/hip_runtime.h>
#include <hip/hip_bf16.h>

// ---------------------------------------------------------------------------
// Deformable PS-RoI pooling head for MI455X (gfx1250, wave32).
//  Stage 1: 3x3 SAME conv as fp32 GEMM on the matrix pipe
//           (v_wmma_f32_16x16x4_f32) with double-buffered async LDS staging
//           (global_load_async_to_lds_b128 / s_wait_asynccnt).
//  Stage 2: PS-RoI pool of offset map -> scaled per-bin offsets
//  Stage 3: deformable PS-RoI pool of features -> [N, 8, 7, 7]
// ---------------------------------------------------------------------------

typedef __attribute__((ext_vector_type(2))) float v2f;
typedef __attribute__((ext_vector_type(8))) float v8f;

#define Hf 128
#define Wf 128
#define Cf 392          // K*K*8 = in channels = out channels
#define CHUNK 56        // channel chunk staged in LDS (392/7, multiple of 4)
#define NCHUNKS 7
#define NTILES 25       // ceil(392/16)
#define KBINS 49
#define FEAT_STRIDE 16.0f
#define LAMDA 0.1f

// ---------------------------------------------------------------------------
// Async staging of one channel chunk of the strip's receptive field:
// [3 rows][18 cols][CHUNK channels] -> LDS buffer, zero-filled at borders.
// In-bounds cells use GLOBAL_LOAD_ASYNC_TO_LDS_B128 (ASYNCcnt); border cells
// are zeroed with a plain LDS store (DScnt, handled by the barrier).
// ---------------------------------------------------------------------------
__device__ __forceinline__ void stage_chunk(const float* __restrict__ feat,
                                            float* __restrict__ buf,
                                            int y, int x0, int cbase, int tid) {
  for (int u = tid; u < 3 * 18 * (CHUNK / 4); u += 256) {
    const int cell = u / (CHUNK / 4);
    const int q    = u - cell * (CHUNK / 4);
    const int r    = cell / 18;
    const int cc   = cell - r * 18;
    const int gy   = y + r - 1;
    const int gx   = x0 + cc - 1;
    float* lp = buf + cell * CHUNK + q * 4;
    if (gy >= 0 && gy < Hf && gx >= 0 && gx < Wf) {
      const float* gp = feat + ((size_t)(gy * Wf + gx) * Cf + cbase + q * 4);
      // Low 32 bits of a flat LDS pointer are the LDS byte address
      const unsigned int ldsAddr = (unsigned int)(size_t)lp;
      asm volatile("global_load_async_to_lds_b128 %0, %1, off"
                   :: "v"(ldsAddr), "v"(gp) : "memory");
    } else {
      *(float4*)lp = make_float4(0.f, 0.f, 0.f, 0.f);  // SAME padding
    }
  }
}

__device__ __forceinline__ void wait_async_all() {
  asm volatile("s_wait_asynccnt 0x0" ::: "memory");
}

// ---------------------------------------------------------------------------
// Stage 1: conv3x3 via WMMA.  grid = (8 x-strips, 128 rows, 4 ntile-groups),
// block = 256 threads (8 waves).  Wave w computes the 16x16 tile
// (16 pixels of the strip) x (out channels [16*nt, 16*nt+16)).
// K = 9*392 reduction split into 7 chunks of 56 channels, double-buffered:
// chunk c+1 streams into LDS asynchronously while chunk c feeds the WMMAs.
// ---------------------------------------------------------------------------
__global__ __launch_bounds__(256) void conv3x3_wmma_f32(
    const float* __restrict__ feat,   // [128,128,392]
    const float* __restrict__ w,      // [3,3,392,392] (HWIO)
    float* __restrict__ out)          // [128,128,392]
{
  __shared__ float lds[2][3 * 18 * CHUNK];   // 2 x 12,096 B

  const int xs   = blockIdx.x;            // 0..7  (16-pixel strip along W)
  const int y    = blockIdx.y;            // 0..127
  const int ng   = blockIdx.z;            // 0..3
  const int tid  = threadIdx.x;
  const int wave = tid >> 5;
  const int lane = tid & 31;

  const int nt = ng * 8 + wave;           // output-channel tile id
  const bool active = (nt < NTILES);      // wave-uniform -> EXEC all-1s inside
  const int n0 = nt * 16;

  const int m  = lane & 15;               // A-row / spatial index within strip
  const int kh = lane >> 4;               // K-half: VGPR0/1 hold K = 2*kh + {0,1}
  const int nC = n0 + (lane & 15);        // B/C column (output channel)
  const int nB = nC < (Cf - 1) ? nC : (Cf - 1);   // clamped for B loads
  const int x0 = xs * 16;

  v8f acc = {};

  // Prologue: stage chunk 0.
  stage_chunk(feat, &lds[0][0], y, x0, 0, tid);
  wait_async_all();
  __syncthreads();

  for (int c = 0; c < NCHUNKS; ++c) {
    // Prefetch next chunk into the other buffer (overlaps the WMMA loop).
    if (c + 1 < NCHUNKS)
      stage_chunk(feat, &lds[(c + 1) & 1][0], y, x0, (c + 1) * CHUNK, tid);

    if (active) {
      const float* buf = &lds[c & 1][0];
      const int cbase = c * CHUNK;
      for (int p = 0; p < 9; ++p) {                 // 3x3 tap position
        const int ky = p / 3, kx = p - ky * 3;
        const float* Ap = buf + ((ky * 18) + m + kx) * CHUNK + 2 * kh;
        const float* Bp = w + ((size_t)(p * Cf + cbase + 2 * kh) * Cf) + nB;
        for (int cb = 0; cb < CHUNK; cb += 4) {     // 14 k-steps of 4
          v2f av, bv;
          av.x = Ap[cb];
          av.y = Ap[cb + 1];
          bv.x = Bp[(size_t)cb * Cf];
          bv.y = Bp[(size_t)(cb + 1) * Cf];
          acc = __builtin_amdgcn_wmma_f32_16x16x4_f32(
              /*neg_a=*/false, av, /*neg_b=*/false, bv,
              /*c_mod=*/(short)0, acc, /*reuse_a=*/false, /*reuse_b=*/false);
        }
      }
    }

    // Next chunk resident + current buffer free for restaging.
    wait_async_all();
    __syncthreads();
  }

  // Store C: lane -> column nC, VGPR r -> spatial row M = r + 8*kh.
  if (active && nC < Cf) {
    const int mg = y * Wf + x0;
#pragma unroll
    for (int r = 0; r < 8; ++r) {
      const int M = r + 8 * kh;
      out[(size_t)(mg + M) * Cf + nC] = acc[r];
    }
  }
}

// ---------------------------------------------------------------------------
// Bilinear sample with reference semantics: corner indices clipped to
// [0, H-1]/[0, W-1], interpolation weights from the *unclipped* coords.
// ---------------------------------------------------------------------------
__device__ __forceinline__ float bilin392(const float* __restrict__ f,
                                          float y, float x, int c) {
  const float y0f = floorf(y), x0f = floorf(x);
  const float wy = y - y0f, wx = x - x0f;
  int y0 = (int)y0f, x0 = (int)x0f;
  int y1 = y0 + 1,   x1 = x0 + 1;
  y0 = min(max(y0, 0), Hf - 1);  y1 = min(max(y1, 0), Hf - 1);
  x0 = min(max(x0, 0), Wf - 1);  x1 = min(max(x1, 0), Wf - 1);
  const float v00 = f[(size_t)(y0 * Wf + x0) * Cf + c];
  const float v01 = f[(size_t)(y0 * Wf + x1) * Cf + c];
  const float v10 = f[(size_t)(y1 * Wf + x0) * Cf + c];
  const float v11 = f[(size_t)(y1 * Wf + x1) * Cf + c];
  return (1.f - wy) * (1.f - wx) * v00 + (1.f - wy) * wx * v01
       + wy * (1.f - wx) * v10 + wy * wx * v11;
}

// ---------------------------------------------------------------------------
// Stage 2: PS-RoI pool offset_map at bin centers, scale by roi size * lamda.
// One thread per (roi, bin, d) with d = g*2 + s; output layout [N, 49, 4, 2].
// ---------------------------------------------------------------------------
__global__ __launch_bounds__(256) void psroi_offsets(
    const float* __restrict__ omap,     // [128,128,392]
    const float* __restrict__ rois,     // [N,5]
    float* __restrict__ off,            // [N,392] == [N,49,4,2]
    int total)
{
  const int idx = blockIdx.x * blockDim.x + threadIdx.x;
  if (idx >= total) return;
  const int n   = idx / Cf;
  const int r   = idx - n * Cf;
  const int bin = r >> 3;               // 0..48
  const int d   = r & 7;                // g*2 + s
  const int s   = d & 1;

  const float* roi = rois + (size_t)n * 5;
  const float x1 = roi[1] / FEAT_STRIDE;
  const float y1 = roi[2] / FEAT_STRIDE;
  const float x2 = (roi[3] + 1.0f) / FEAT_STRIDE;
  const float y2 = (roi[4] + 1.0f) / FEAT_STRIDE;
  const float bw = (x2 - x1) * (1.0f / 7.0f);
  const float bh = (y2 - y1) * (1.0f / 7.0f);
  const int i = bin / 7, j = bin - i * 7;
  const float cx = x1 + ((float)j + 0.5f) * bw;
  const float cy = y1 + ((float)i + 0.5f) * bh;

  const float v = bilin392(omap, cy, cx, bin * 8 + d);
  const float scale = (s == 0 ? (roi[3] - roi[1] + 1.0f)
                              : (roi[4] - roi[2] + 1.0f)) * LAMDA;
  off[idx] = v * scale;
}

// ---------------------------------------------------------------------------
// Stage 3: deformable PS-RoI pool of the features with predicted offsets.
// One thread per output element out[n, d, i, j] (d = g*2 + dd, bin = i*7+j).
// ---------------------------------------------------------------------------
__global__ __launch_bounds__(256) void psroi_deform(
    const float* __restrict__ feat,     // [128,128,392]
    const float* __restrict__ rois,     // [N,5]
    const float* __restrict__ off,      // [N,49,4,2]
    float* __restrict__ out,            // [N,8,7,7]
    int total)
{
  const int o = blockIdx.x * blockDim.x + threadIdx.x;
  if (o >= total) return;
  const int n  = o / (8 * KBINS);
  const int r  = o - n * (8 * KBINS);
  const int d  = r / KBINS;             // 0..7
  const int bin = r - d * KBINS;        // i*7 + j
  const int i = bin / 7, j = bin - i * 7;
  const int g = d >> 1;                 // channel group 0..3

  const float* roi = rois + (size_t)n * 5;
  const float x1 = roi[1] / FEAT_STRIDE;
  const float y1 = roi[2] / FEAT_STRIDE;
  const float x2 = (roi[3] + 1.0f) / FEAT_STRIDE;
  const float y2 = (roi[4] + 1.0f) / FEAT_STRIDE;
  const float bw = (x2 - x1) * (1.0f / 7.0f);
  const float bh = (y2 - y1) * (1.0f / 7.0f);

  const float offx = off[(size_t)n * Cf + bin * 8 + g * 2 + 0];
  const float offy = off[(size_t)n * Cf + bin * 8 + g * 2 + 1];
  const float cx = x1 + ((float)j + 0.5f) * bw + offx / FEAT_STRIDE;
  const float cy = y1 + ((float)i + 0.5f) * bh + offy / FEAT_STRIDE;

  out[o] = bilin392(feat, cy, cx, bin * 8 + d);
}

// ---------------------------------------------------------------------------
extern "C" void kernel_launch(void* const* d_in, const int* in_sizes, int n_in,
                              void* d_out, int out_size, void* d_ws, size_t ws_size,
                              hipStream_t stream) {
  const float* features = (const float*)d_in[0];   // [1,128,128,392]
  const float* rois     = (const float*)d_in[1];   // [N,5]
  const float* conv_w   = (const float*)d_in[2];   // [3,3,392,392]
  float* out = (float*)d_out;                      // [N,8,7,7]

  const int N = in_sizes[1] / 5;

  float* omap = (float*)d_ws;                      // [128,128,392]  ~25.7 MB
  float* off  = omap + (size_t)Hf * Wf * Cf;       // [N,392]        ~3.2 MB

  // Stage 1: 3x3 conv on the matrix pipe.
  dim3 gconv(8, Hf, 4);
  conv3x3_wmma_f32<<<gconv, 256, 0, stream>>>(features, conv_w, omap);

  // Stage 2: predicted offsets.
  const int total = N * Cf;                        // N*49*8 == N*8*49
  psroi_offsets<<<(total + 255) / 256, 256, 0, stream>>>(omap, rois, off, total);

  // Stage 3: deformable pooling.
  psroi_deform<<<(total + 255) / 256, 256, 0, stream>>>(features, rois, off, out, total);
}